// Loss_49185965474290
// MI455X (gfx1250) — compile-verified
//
#include <hip/hip_runtime.h>
#include <hip/hip_bf16.h>
#include <math.h>

typedef __attribute__((ext_vector_type(2))) float v2f;
typedef __attribute__((ext_vector_type(8))) float v8f;

#define WAVE 32

// ---------------- workspace layout (floats) ----------------
// [0..3]   inter[c]   [4..7] denom_p[c]   [8..11] denom_o[c]
// [12] ce_sum  [13] focal_sum  [14] chamfer_acc  [15] lap_acc
// [32..]            deg  : 6*4096
// [32+6*4096..]     nsum : 6*4096*3
#define WS_ACC   0
#define WS_DEG   32
#define WS_NSUM  (32 + 6 * 4096)
#define WS_TOTAL (32 + 6 * 4096 + 6 * 4096 * 3)

__device__ __forceinline__ float wave_sum(float v) {
  #pragma unroll
  for (int off = 16; off > 0; off >>= 1) v += __shfl_down(v, off, WAVE);
  return v;
}

__device__ __forceinline__ float logsig(float x) {
  return fminf(x, 0.0f) - log1pf(expf(-fabsf(x)));
}

// ---------------- zero workspace ----------------
__global__ void zero_ws_kernel(float* w, int n) {
  int i = blockIdx.x * blockDim.x + threadIdx.x;
  if (i < n) w[i] = 0.0f;
}

// ---------------- dice + CE over [1,4,128,128,128] ----------------
// 8 voxels per thread (grid-strided, coalesced) to amortize the wave
// reduction + atomics; the kernel is a pure 40MB HBM stream otherwise.
__global__ void dice_ce_kernel(const float* __restrict__ logits,
                               const float* __restrict__ label,
                               float* __restrict__ acc) {
  const int CS = 128 * 128 * 128;       // channel stride
  const int NT = CS / 8;                // total threads = 262144
  int tid = blockIdx.x * blockDim.x + threadIdx.x;

  float dp[4] = {0, 0, 0, 0}, din[4] = {0, 0, 0, 0}, dn[4] = {0, 0, 0, 0};
  float ce = 0.0f;

  #pragma unroll 2
  for (int j = 0; j < 8; ++j) {
    int idx = tid + j * NT;
    if (j + 1 < 8) {
      __builtin_prefetch(&logits[idx + NT], 0, 3);
      __builtin_prefetch(&label[idx + NT], 0, 3);
    }
    float l0 = logits[idx];
    float l1 = logits[idx + CS];
    float l2 = logits[idx + 2 * CS];
    float l3 = logits[idx + 3 * CS];
    float m = fmaxf(fmaxf(l0, l1), fmaxf(l2, l3));
    float e0 = expf(l0 - m), e1 = expf(l1 - m), e2 = expf(l2 - m), e3 = expf(l3 - m);
    float se = e0 + e1 + e2 + e3;
    float inv = 1.0f / se;
    float p[4] = {e0 * inv, e1 * inv, e2 * inv, e3 * inv};
    float lse = logf(se);

    int lbl = (int)label[idx];
    lbl = lbl < 0 ? 0 : (lbl > 3 ? 3 : lbl);

    dp[0] += p[0]; dp[1] += p[1]; dp[2] += p[2]; dp[3] += p[3];
    din[lbl] += p[lbl];
    dn[lbl] += 1.0f;
    float lps[4] = {l0 - m - lse, l1 - m - lse, l2 - m - lse, l3 - m - lse};
    ce -= lps[lbl];
  }

  int lane = threadIdx.x & (WAVE - 1);
  #pragma unroll
  for (int c = 0; c < 4; ++c) {
    float s;
    s = wave_sum(din[c]); if (lane == 0) atomicAdd(&acc[WS_ACC + c], s);
    s = wave_sum(dp[c]);  if (lane == 0) atomicAdd(&acc[WS_ACC + 4 + c], s);
    s = wave_sum(dn[c]);  if (lane == 0) atomicAdd(&acc[WS_ACC + 8 + c], s);
  }
  float cs = wave_sum(ce);
  if (lane == 0) atomicAdd(&acc[WS_ACC + 12], cs);
}

// ---------------- trilinear sample + sigmoid focal ----------------
__device__ __forceinline__ void trilin_prep(float gx, float gy, float gz,
                                            int idx8[8], float w8[8]) {
  const int W = 128, H = 128, D = 128;
  float x = (gx + 1.0f) * 0.5f * (W - 1);
  float y = (gy + 1.0f) * 0.5f * (H - 1);
  float z = (gz + 1.0f) * 0.5f * (D - 1);
  float x0 = fminf(fmaxf(floorf(x), 0.0f), W - 1.0f);
  float y0 = fminf(fmaxf(floorf(y), 0.0f), H - 1.0f);
  float z0 = fminf(fmaxf(floorf(z), 0.0f), D - 1.0f);
  float x1 = fminf(x0 + 1.0f, W - 1.0f);
  float y1 = fminf(y0 + 1.0f, H - 1.0f);
  float z1 = fminf(z0 + 1.0f, D - 1.0f);
  float fx = x - x0, fy = y - y0, fz = z - z0;
  int xi0 = (int)x0, xi1 = (int)x1, yi0 = (int)y0, yi1 = (int)y1, zi0 = (int)z0, zi1 = (int)z1;
  const int SY = 128, SZ = 128 * 128;
  idx8[0] = zi0 * SZ + yi0 * SY + xi0; w8[0] = (1 - fx) * (1 - fy) * (1 - fz);
  idx8[1] = zi0 * SZ + yi0 * SY + xi1; w8[1] = fx * (1 - fy) * (1 - fz);
  idx8[2] = zi0 * SZ + yi1 * SY + xi0; w8[2] = (1 - fx) * fy * (1 - fz);
  idx8[3] = zi0 * SZ + yi1 * SY + xi1; w8[3] = fx * fy * (1 - fz);
  idx8[4] = zi1 * SZ + yi0 * SY + xi0; w8[4] = (1 - fx) * (1 - fy) * fz;
  idx8[5] = zi1 * SZ + yi0 * SY + xi1; w8[5] = fx * (1 - fy) * fz;
  idx8[6] = zi1 * SZ + yi1 * SY + xi0; w8[6] = (1 - fx) * fy * fz;
  idx8[7] = zi1 * SZ + yi1 * SY + xi1; w8[7] = fx * fy * fz;
}

__global__ void focal_kernel(const float* __restrict__ logits,
                             const float* __restrict__ label,
                             const float* __restrict__ vertices,
                             float* __restrict__ acc) {
  const int CS = 128 * 128 * 128;
  int n = blockIdx.x * blockDim.x + threadIdx.x;  // 8192 threads
  float gx = 2.0f * vertices[n * 3 + 0] - 1.0f;
  float gy = 2.0f * vertices[n * 3 + 1] - 1.0f;
  float gz = 2.0f * vertices[n * 3 + 2] - 1.0f;
  int idx8[8]; float w8[8];
  trilin_prep(gx, gy, gz, idx8, w8);

  float lv[4] = {0, 0, 0, 0};
  float tv = 0.0f;
  #pragma unroll
  for (int j = 0; j < 8; ++j) {
    float w = w8[j]; int id = idx8[j];
    lv[0] += w * logits[id];
    lv[1] += w * logits[id + CS];
    lv[2] += w * logits[id + 2 * CS];
    lv[3] += w * logits[id + 3 * CS];
    tv += w * label[id];
  }
  int tidx = (int)tv;
  tidx = tidx < 0 ? 0 : (tidx > 3 ? 3 : tidx);

  float loss = 0.0f;
  #pragma unroll
  for (int c = 0; c < 4; ++c) {
    float x = lv[c];
    float t = (c == tidx) ? 1.0f : 0.0f;
    float p = 1.0f / (1.0f + expf(-x));
    float pt = t * p + (1.0f - t) * (1.0f - p);
    float log_pt = t * logsig(x) + (1.0f - t) * logsig(-x);
    float om = 1.0f - pt;
    loss += -(om * om) * log_pt;  // gamma = 2
  }
  float s = wave_sum(loss);
  if ((threadIdx.x & (WAVE - 1)) == 0) atomicAdd(&acc[WS_ACC + 13], s);
}

// ---------------- chamfer via v_wmma_f32_16x16x4_f32 ----------------
// Norm-folded WMMA: A = (-2x,-2y,-2z, 1), B = (bx,by,bz,|b|^2)
//   => D[M][N] = |b_N|^2 - 2 a_M.b_N  (min over N, add |a_M|^2 at the end).
// Y (48KB) is staged into LDS once per workgroup with the gfx1250 async
// global->LDS path; the inner loop is ds_load + wmma + 8x v_min only.
__global__ void __launch_bounds__(128)
chamfer_kernel(const float* __restrict__ pverts,
               const float* __restrict__ lpoints,
               float* __restrict__ acc) {
  const int V = 4096, P = 4096;
  __shared__ float Ys[4096 * 3];        // 48KB

  int job = blockIdx.y;                 // 0..11
  int dir = job & 1;
  int pair = job >> 1;                  // i*3 + k
  int k = pair % 3;
  const float* Xp = pverts + (size_t)pair * V * 3;
  const float* Yp = lpoints + (size_t)k * P * 3;
  const float* X = dir ? Yp : Xp;
  const float* Y = dir ? Xp : Yp;

  // ---- async stage Y -> LDS (49152 B = 3072 x b128, 128 threads x 24) ----
  {
    unsigned ldsbase = (unsigned)(size_t)&Ys[0];
    #pragma unroll 4
    for (int j = 0; j < 24; ++j) {
      int off = (j * 128 + (int)threadIdx.x) * 16;
      unsigned lds = ldsbase + (unsigned)off;
      unsigned long long g = (unsigned long long)(const char*)Y + (unsigned)off;
      asm volatile("global_load_async_to_lds_b128 %0, %1, off"
                   :: "v"(lds), "v"(g) : "memory");
    }
    asm volatile("s_wait_asynccnt 0x0" ::: "memory");
    __syncthreads();
  }

  int wave = threadIdx.x >> 5;          // 4 waves / block
  int lane = threadIdx.x & 31;
  int m = lane & 15;
  bool hi = lane >= 16;
  int rowbase = (blockIdx.x * 4 + wave) * 16;

  // A tile (16x4): lanes 0-15 hold K=0,1; lanes 16-31 hold K=2,3
  int arow = rowbase + m;
  float ax = X[arow * 3 + 0];
  float ay = X[arow * 3 + 1];
  float az = X[arow * 3 + 2];
  float na_full = ax * ax + ay * ay + az * az;
  v2f a;
  a.x = hi ? (-2.0f * az) : (-2.0f * ax);
  a.y = hi ? 1.0f : (-2.0f * ay);

  float minv[8];
  #pragma unroll
  for (int r = 0; r < 8; ++r) minv[r] = 3.402823e38f;

  for (int ct = 0; ct < P / 16; ++ct) {
    int col = ct * 16 + m;
    float bx = Ys[col * 3 + 0];
    float by = Ys[col * 3 + 1];
    float bz = Ys[col * 3 + 2];
    float nb = bx * bx + by * by + bz * bz;
    v2f b;
    b.x = hi ? bz : bx;
    b.y = hi ? nb : by;
    v8f c = {};
    v8f d = __builtin_amdgcn_wmma_f32_16x16x4_f32(
        false, a, false, b, (short)0, c, false, false);
    #pragma unroll
    for (int r = 0; r < 8; ++r) minv[r] = fminf(minv[r], d[r]);
  }

  // min across the 16 lanes of each half (rows r and 8+r of the strip)
  #pragma unroll
  for (int r = 0; r < 8; ++r) {
    float v = minv[r];
    v = fminf(v, __shfl_xor(v, 1, WAVE));
    v = fminf(v, __shfl_xor(v, 2, WAVE));
    v = fminf(v, __shfl_xor(v, 4, WAVE));
    v = fminf(v, __shfl_xor(v, 8, WAVE));
    minv[r] = v;
  }

  int mbase = hi ? 8 : 0;
  float s = 0.0f;
  #pragma unroll
  for (int r = 0; r < 8; ++r)
    s += minv[r] + __shfl(na_full, mbase + r, WAVE);  // + |a_row|^2
  if (m == 0) atomicAdd(&acc[WS_ACC + 14], s);        // lanes 0 and 16
}

// ---------------- laplacian: scatter then finalize ----------------
__global__ void lap_scatter_kernel(const float* __restrict__ pverts,
                                   const int* __restrict__ pedges,
                                   float* __restrict__ ws) {
  const int V = 4096, E = 12288;
  int id = blockIdx.x * blockDim.x + threadIdx.x;  // 6*E threads
  if (id >= 6 * E) return;
  int mesh = id / E;          // i*3+k
  int e = id % E;
  int k = mesh % 3;
  int v0 = pedges[((size_t)k * E + e) * 2 + 0];
  int v1 = pedges[((size_t)k * E + e) * 2 + 1];
  const float* vv = pverts + (size_t)mesh * V * 3;
  float* deg = ws + WS_DEG + (size_t)mesh * V;
  float* ns = ws + WS_NSUM + (size_t)mesh * V * 3;
  atomicAdd(&deg[v0], 1.0f);
  atomicAdd(&deg[v1], 1.0f);
  #pragma unroll
  for (int d = 0; d < 3; ++d) {
    atomicAdd(&ns[v0 * 3 + d], vv[v1 * 3 + d]);
    atomicAdd(&ns[v1 * 3 + d], vv[v0 * 3 + d]);
  }
}

__global__ void lap_finalize_kernel(const float* __restrict__ pverts,
                                    float* __restrict__ ws) {
  const int V = 4096;
  int id = blockIdx.x * blockDim.x + threadIdx.x;  // 6*V threads
  int mesh = id / V;
  int v = id % V;
  const float* vv = pverts + (size_t)mesh * V * 3;
  const float* deg = ws + WS_DEG + (size_t)mesh * V;
  const float* ns = ws + WS_NSUM + (size_t)mesh * V * 3;
  float dg = fmaxf(deg[v], 1.0f);
  float lx = ns[v * 3 + 0] / dg - vv[v * 3 + 0];
  float ly = ns[v * 3 + 1] / dg - vv[v * 3 + 1];
  float lz = ns[v * 3 + 2] / dg - vv[v * 3 + 2];
  float nrm = sqrtf(lx * lx + ly * ly + lz * lz);
  float s = wave_sum(nrm);
  if ((threadIdx.x & (WAVE - 1)) == 0)
    atomicAdd(&ws[WS_ACC + 15], s * (1.0f / V));  // accumulates sum of per-mesh means
}

// ---------------- final combine ----------------
__global__ void combine_kernel(const float* __restrict__ acc, float* __restrict__ out) {
  if (threadIdx.x == 0 && blockIdx.x == 0) {
    const float EPS = 1e-5f;
    float dice = 0.0f;
    #pragma unroll
    for (int c = 0; c < 4; ++c) {
      float inter = acc[WS_ACC + c];
      float denom = acc[WS_ACC + 4 + c] + acc[WS_ACC + 8 + c];
      dice += 1.0f - (2.0f * inter + EPS) / (denom + EPS);
    }
    dice *= 0.25f;
    float ce = acc[WS_ACC + 12] * (1.0f / 2097152.0f);
    float inout = acc[WS_ACC + 13] * (1.0f / (4.0f * 8192.0f));
    float cham = acc[WS_ACC + 14] * (1.0f / (6.0f * 4096.0f * 4096.0f));
    float lap = 0.05f * acc[WS_ACC + 15] * (1.0f / 6.0f);
    out[0] = (dice + ce) + 10.0f * inout + (cham + lap);
  }
}

extern "C" void kernel_launch(void* const* d_in, const int* in_sizes, int n_in,
                              void* d_out, int out_size, void* d_ws, size_t ws_size,
                              hipStream_t stream) {
  const float* logit_map = (const float*)d_in[0];  // [1,4,128,128,128]
  const float* label     = (const float*)d_in[1];  // [1,1,128,128,128]
  const float* vertices  = (const float*)d_in[2];  // [8192,3]
  const float* pverts    = (const float*)d_in[3];  // [2,3,4096,3]
  const int*   pedges    = (const int*)d_in[4];    // [3,12288,2]
  const float* lpoints   = (const float*)d_in[5];  // [3,4096,3]
  float* out = (float*)d_out;
  float* ws = (float*)d_ws;

  (void)in_sizes; (void)n_in; (void)out_size; (void)ws_size;

  // 1) zero accumulators + laplacian scratch
  {
    int n = WS_TOTAL;
    zero_ws_kernel<<<(n + 255) / 256, 256, 0, stream>>>(ws, n);
  }
  // 2) dice + CE: 262144 threads x 8 voxels, HBM-bound stream
  dice_ce_kernel<<<262144 / 256, 256, 0, stream>>>(logit_map, label, ws);
  // 3) trilinear + focal (8192 points)
  focal_kernel<<<8192 / 256, 256, 0, stream>>>(logit_map, label, vertices, ws);
  // 4) chamfer: 12 directed jobs, 256 row-strips each, 4 waves/block
  {
    dim3 grid(64, 12, 1);   // 64 blocks * 4 waves = 256 strips of 16 rows
    chamfer_kernel<<<grid, 128, 0, stream>>>(pverts, lpoints, ws);
  }
  // 5) laplacian scatter (6 meshes x 12288 edges)
  {
    int n = 6 * 12288;
    lap_scatter_kernel<<<(n + 255) / 256, 256, 0, stream>>>(pverts, pedges, ws);
  }
  // 6) laplacian finalize (6 x 4096 vertices)
  lap_finalize_kernel<<<6 * 4096 / 256, 256, 0, stream>>>(pverts, ws);
  // 7) combine into scalar
  combine_kernel<<<1, 32, 0, stream>>>(ws, out);
}